// APKDA_20718922235948
// MI455X (gfx1250) — compile-verified
//
#include <hip/hip_runtime.h>
#include <math.h>

typedef float v2f __attribute__((ext_vector_type(2)));
typedef float v8f __attribute__((ext_vector_type(8)));

#define BATCH   256
#define FEAT    512
#define HW      49
#define B2      512
#define NUM_ID  32
#define GROUP   16

// ---------------------------------------------------------------------------
// Kernel 1: mean-pool over 7x7, L2-normalize each row, write interleaved f.
// One block per source row (512 rows), each thread owns 2 features.
// Bandwidth-dominant stage: streams 51 MB once.
// ---------------------------------------------------------------------------
__global__ __launch_bounds__(256) void apkda_pool_norm(
    const float* __restrict__ outputs, const float* __restrict__ targets,
    float* __restrict__ f)
{
    __shared__ float red[256];
    const int r = blockIdx.x;            // 0..255 outputs, 256..511 targets
    const int b = r & 255;
    const float* __restrict__ src  = (r < BATCH) ? outputs : targets;
    const float* __restrict__ base = src + (size_t)b * FEAT * HW;
    const int t = threadIdx.x;

    const float* p0 = base + (size_t)t * HW;
    const float* p1 = base + (size_t)(t + 256) * HW;
    float s0 = 0.0f, s1 = 0.0f;
    #pragma unroll
    for (int k = 0; k < HW; ++k) { s0 += p0[k]; s1 += p1[k]; }
    const float v0 = s0 * (1.0f / HW);
    const float v1 = s1 * (1.0f / HW);

    red[t] = v0 * v0 + v1 * v1;
    __syncthreads();
    #pragma unroll
    for (int s = 128; s > 0; s >>= 1) {
        if (t < s) red[t] += red[t + s];
        __syncthreads();
    }
    const float inv = 1.0f / fmaxf(sqrtf(red[0]), 1e-12f);

    // interleave: identity c = b>>3, slot p = b&7; targets go to +8
    const int c = b >> 3, p = b & 7;
    const int drow = c * GROUP + p + ((r < BATCH) ? 0 : 8);
    f[(size_t)drow * FEAT + t]       = v0 * inv;
    f[(size_t)drow * FEAT + t + 256] = v1 * inv;
}

// ---------------------------------------------------------------------------
// Kernel 2: sim = f @ f^T via V_WMMA_F32_16X16X4_F32 (full fp32 precision).
// One wave (32 lanes) per 16x16 output tile; 4 waves per block.
// A-frag layout (16x4 f32): lane m=lane&15, VGPR0->K=2*kh, VGPR1->K=2*kh+1,
// kh = lane>>4. B-frag (4x16) mirrors it with n=lane&15.
// ---------------------------------------------------------------------------
__global__ __launch_bounds__(128) void apkda_gemm_ffT(
    const float* __restrict__ f, float* __restrict__ sim)
{
    const int lane = threadIdx.x & 31;
    const int wave = threadIdx.x >> 5;
    const int ti = blockIdx.y;                 // row tile 0..31
    const int tj = blockIdx.x * 4 + wave;      // col tile 0..31
    const int mn = lane & 15;
    const int kh = lane >> 4;                  // selects K-pair {2kh, 2kh+1}

    const float* __restrict__ arow = f + (size_t)(ti * 16 + mn) * FEAT;
    const float* __restrict__ brow = f + (size_t)(tj * 16 + mn) * FEAT;

    v8f acc = {};
    for (int kb = 0; kb < FEAT; kb += 4) {
        v2f a = *(const v2f*)(arow + kb + kh * 2);   // 8B-aligned global_load_b64
        v2f b = *(const v2f*)(brow + kb + kh * 2);
        acc = __builtin_amdgcn_wmma_f32_16x16x4_f32(
            /*neg_a=*/false, a, /*neg_b=*/false, b,
            /*c_mod=*/(short)0, acc, /*reuse_a=*/false, /*reuse_b=*/false);
    }
    // C/D layout: lane -> n = lane&15; VGPR r -> m = r + 8*(lane>>4)
    #pragma unroll
    for (int rr = 0; rr < 8; ++rr) {
        const int m = rr + kh * 8;
        sim[(size_t)(ti * 16 + m) * B2 + tj * 16 + mn] = acc[rr];
    }
}

// ---------------------------------------------------------------------------
// Kernel 3: per-identity ranking. Block c stages rows [16c,16c+16) of sim
// (16x512 = 32 KB LDS), then thread (i,j) computes
//   rk_all = 1 + sum_k relu(S[i][k] - S[i][16c+j])      (mask term is 0 anyway)
//   rk_pos = 1 + sum_{k<16} relu(S[i][16c+k] - S[i][16c+j])
// and contributes rk_pos/rk_all/8192. Fixed-order tree reduce -> partial[c].
// ---------------------------------------------------------------------------
__global__ __launch_bounds__(256) void apkda_rank(
    const float* __restrict__ sim, float* __restrict__ partial)
{
    __shared__ float S[GROUP][B2];     // 32 KB
    __shared__ float red[256];
    const int cblk = blockIdx.x;
    const int t = threadIdx.x;

    // cooperative float4 load of 16 contiguous rows
    const float4* __restrict__ src = (const float4*)(sim + (size_t)cblk * GROUP * B2);
    float4* dst = (float4*)(&S[0][0]);
    #pragma unroll
    for (int l = t; l < GROUP * B2 / 4; l += 256) dst[l] = src[l];
    __syncthreads();

    const int i = t >> 4, j = t & 15;
    const float aij = S[i][cblk * GROUP + j];

    float rall = 1.0f;
    for (int k = 0; k < B2; ++k)
        rall += fmaxf(S[i][k] - aij, 0.0f);

    float rpos = 1.0f;
    #pragma unroll
    for (int k = 0; k < GROUP; ++k)
        rpos += fmaxf(S[i][cblk * GROUP + k] - aij, 0.0f);

    red[t] = (rpos / rall) * (1.0f / (GROUP * B2));
    __syncthreads();
    #pragma unroll
    for (int s = 128; s > 0; s >>= 1) {
        if (t < s) red[t] += red[t + s];
        __syncthreads();
    }
    if (t == 0) partial[cblk] = red[0];
}

// ---------------------------------------------------------------------------
// Kernel 4: single wave32 reduction of 32 partials -> out = 1 - ap.
// ---------------------------------------------------------------------------
__global__ __launch_bounds__(32) void apkda_finalize(
    const float* __restrict__ partial, float* __restrict__ out)
{
    const int lane = threadIdx.x;
    float v = partial[lane];
    #pragma unroll
    for (int off = 16; off > 0; off >>= 1)
        v += __shfl_down(v, off, 32);
    if (lane == 0) out[0] = 1.0f - v;
}

extern "C" void kernel_launch(void* const* d_in, const int* in_sizes, int n_in,
                              void* d_out, int out_size, void* d_ws, size_t ws_size,
                              hipStream_t stream) {
    const float* outputs = (const float*)d_in[0];
    const float* targets = (const float*)d_in[1];
    float* out = (float*)d_out;

    float* f       = (float*)d_ws;               // 512*512 f32 = 1 MB
    float* sim     = f + (size_t)B2 * FEAT;      // 512*512 f32 = 1 MB
    float* partial = sim + (size_t)B2 * B2;      // 32 f32

    apkda_pool_norm<<<2 * BATCH, 256, 0, stream>>>(outputs, targets, f);
    apkda_gemm_ffT<<<dim3(8, 32), 128, 0, stream>>>(f, sim);
    apkda_rank<<<NUM_ID, 256, 0, stream>>>(sim, partial);
    apkda_finalize<<<1, 32, 0, stream>>>(partial, out);
}